// CheapTrick_12584254177890
// MI455X (gfx1250) — compile-verified
//
#include <hip/hip_runtime.h>
#include <math.h>

// ---------------- CheapTrick on MI455X (gfx1250) ----------------
// One workgroup (= 1 wave32) per frame; all per-frame state in LDS
// (~50KB of 320KB/WGP). Each 1024-pt FFT = four-step 32x32 decomposition,
// X = F32 * (T o (A * F32)); each complex 32x32 matmul = 128
// V_WMMA_F32_16X16X4_F32 ops. Helpers take LDS arrays by reference with
// compile-time strides so all accesses stay addrspace(3) (pure ds_load /
// ds_store with 32-bit addressing, no generic-pointer null checks).

#define FSr     16000.0f
#define FFTN    1024
#define HALFN   512
#define DELTA   (FSr / (float)FFTN)      // 15.625
#define NFRAMES 4096
#define XLEN    336000
#define Q1C     (-0.15f)
#define PI_F    3.14159265358979323846f

typedef __attribute__((ext_vector_type(2))) float v2f;
typedef __attribute__((ext_vector_type(8))) float v8f;

__device__ __forceinline__ v8f wmma4(v2f a, v2f b, v8f c) {
  // D(16x16,f32) = A(16x4,f32) * B(4x16,f32) + C
  return __builtin_amdgcn_wmma_f32_16x16x4_f32(
      /*neg_a=*/false, a, /*neg_b=*/false, b,
      /*c_mod=*/(short)0, c, /*reuse_a=*/false, /*reuse_b=*/false);
}

// Complex 32x32 matmul D = (Ar + i Ai) * (Br + i Bi), D row-major in LDS.
// A indexed as Ar[m*ARS + k*ACS], B as Br[k*BRS + n*BCS] (template consts).
// Fragment layouts per CDNA5 ISA 7.12.2:
//   A 16x4 : lanes 0-15 -> M=lane, K={k0,k0+1}; lanes 16-31 -> M=lane-16, K={k0+2,k0+3}
//   B 4x16 : mirrored half-wave split over K, N striped across lanes
//   C 16x16: VGPR v -> M = v (lanes 0-15) / v+8 (lanes 16-31), N = lane&15
template <int ARS, int ACS, int BRS, int BCS>
__device__ __attribute__((always_inline)) inline
void cmatmul32(int lane,
               const float (&Ar)[1024], const float (&Ai)[1024],
               const float (&Br)[1024], const float (&Bi)[1024],
               float (&Dr)[1024], float (&Di)[1024]) {
  const int half2 = (lane >> 4) << 1;   // 0 or 2
  const int l15   = lane & 15;
#pragma unroll 1
  for (int tm = 0; tm < 32; tm += 16) {
#pragma unroll 1
    for (int tn = 0; tn < 32; tn += 16) {
      v8f accr = {}; v8f acci = {};
      const int m = tm + l15;
      const int n = tn + l15;
#pragma unroll
      for (int k0 = 0; k0 < 32; k0 += 4) {
        const int ka = k0 + half2;
        v2f ar, ai, br, bi, nai;
        ar.x = Ar[m * ARS + ka * ACS];       ar.y = Ar[m * ARS + (ka + 1) * ACS];
        ai.x = Ai[m * ARS + ka * ACS];       ai.y = Ai[m * ARS + (ka + 1) * ACS];
        br.x = Br[ka * BRS + n * BCS];       br.y = Br[(ka + 1) * BRS + n * BCS];
        bi.x = Bi[ka * BRS + n * BCS];       bi.y = Bi[(ka + 1) * BRS + n * BCS];
        nai.x = -ai.x; nai.y = -ai.y;        // f32 WMMA has no A/B neg modifier
        accr = wmma4(ar,  br, accr);         // Re += Ar*Br
        accr = wmma4(nai, bi, accr);         // Re -= Ai*Bi
        acci = wmma4(ar,  bi, acci);         // Im += Ar*Bi
        acci = wmma4(ai,  br, acci);         // Im += Ai*Br
      }
      const int rbase = tm + ((lane >> 4) << 3);
      const int col   = tn + l15;
#pragma unroll
      for (int v = 0; v < 8; ++v) {
        Dr[(rbase + v) * 32 + col] = accr[v];
        Di[(rbase + v) * 32 + col] = acci[v];
      }
    }
  }
}

// 1024-pt complex FFT, four-step 32x32 decomposition.
// s = -1: forward (w = e^{-2pi i/N}); s = +1: inverse direction (1/N applied by caller).
// Input linear; A[n1][n2] = x[n1 + 32*n2] -> stage-1 A strides (1,32).
// Output is naturally linear: Y[k1*32+k2] = X[32*k1 + k2].
__device__ __attribute__((always_inline)) inline
void fft1024(int lane,
             const float (&xr)[1024], const float (&xi)[1024],
             float (&yr)[1024], float (&yi)[1024],
             float (&wr)[1024], float (&wi)[1024],
             const float (&C32)[1024], const float (&S32)[1024], float (&Fi)[1024],
             float s) {
  for (int i = lane; i < 1024; i += 32) Fi[i] = s * S32[i];
  __syncthreads();
  // stage 1: W[n1][k2] = sum_{n2} x[n1+32 n2] * F32[n2][k2]
  cmatmul32<1, 32, 32, 1>(lane, xr, xi, C32, Fi, wr, wi);
  __syncthreads();
  // stage 2: twiddle  W'[r][c] = W[r][c] * e^{s i 2pi r c / 1024}
  for (int i = lane; i < 1024; i += 32) {
    const int r = i >> 5, c = i & 31;
    const float ang = s * (2.0f * PI_F / 1024.0f) * (float)(r * c);
    const float cw = cosf(ang), sw = sinf(ang);
    const float a = wr[i], b = wi[i];
    wr[i] = a * cw - b * sw;
    wi[i] = a * sw + b * cw;
  }
  __syncthreads();
  // stage 3: Y = F32 * W'
  cmatmul32<32, 1, 32, 1>(lane, C32, Fi, wr, wi, yr, yi);
  __syncthreads();
}

__device__ __forceinline__ float wave_sum(float v) {
  for (int m = 16; m >= 1; m >>= 1) v += __shfl_xor(v, m, 32);
  return v;
}

__device__ __forceinline__ float interp_ds(const float (&ds)[2048], float x0d, float xiv) {
  const float hi = x0d + 2047.0f * DELTA;
  float xc = fminf(fmaxf(xiv, x0d), hi);
  float p  = (xc - x0d) / DELTA;
  int b = (int)floorf(p); b = min(max(b, 0), 2047);
  const float frac = p - (float)b;
  const float yb = ds[b];
  const float dy = (b == 2047) ? 0.0f : ds[b + 1] - yb;
  return yb + dy * frac;
}

__global__ __launch_bounds__(32)
void cheaptrick_kernel(const float* __restrict__ x,
                       const float* __restrict__ f0_in,
                       const float* __restrict__ tpos,
                       float* __restrict__ out) {
  __shared__ float C32[1024], S32[1024], Fi[1024];
  __shared__ float xr[1024], xi[1024];
  __shared__ float yr[1024], yi[1024];
  __shared__ float wr[1024], wi[1024];
  __shared__ float ps[1024];
  __shared__ float dsb[2048];
  __shared__ float smb[HALFN + 4];

  const int frame = blockIdx.x;
  const int lane  = threadIdx.x;

  // 32-pt DFT matrix (symmetric): C32[r*32+c]=cos(2pi r c/32), S32=sin(...)
  for (int i = lane; i < 1024; i += 32) {
    const int r = i >> 5, c = i & 31;
    const float ang = (2.0f * PI_F / 32.0f) * (float)(r * c);
    C32[i] = cosf(ang);
    S32[i] = sinf(ang);
  }

  float f0 = f0_in[frame];
  if (f0 < FSr / (float)FFTN) f0 = 500.0f;
  const float pos = tpos[frame];

  // ---- pitch-synchronous windowed segment ----
  const int hw     = (int)floorf(1.5f * FSr / f0 + 0.5f);
  const int center = (int)floorf(pos * FSr + 0.501f) + 1;
  __builtin_prefetch(&x[min(max(center - hw, 1), XLEN) - 1], 0, 0);

  float sw2 = 0.0f, ssw = 0.0f, sws = 0.0f;
  for (int j = lane; j < 1024; j += 32) {
    const float maskf = (j <= 2 * hw) ? 1.0f : 0.0f;
    int idx = center + (j - hw);
    idx = min(max(idx, 1), XLEN) - 1;
    const float sv = x[idx];
    const float wv = (0.5f * cosf(PI_F * (float)(j - hw) / FSr / 1.5f * f0) + 0.5f) * maskf;
    sw2 += wv * wv; ssw += sv * wv; sws += wv;
    yr[j] = sv;     // stash segment
    yi[j] = wv;     // stash window
  }
  sw2 = wave_sum(sw2); ssw = wave_sum(ssw); sws = wave_sum(sws);
  const float invn  = rsqrtf(sw2);
  const float ratio = ssw / sws;        // window normalization cancels in ratio
  __syncthreads();
  for (int j = lane; j < 1024; j += 32) {
    xr[j] = yi[j] * invn * (yr[j] - ratio);
    xi[j] = 0.0f;
  }
  __syncthreads();

  // ---- FFT #1 -> power spectrum ----
  fft1024(lane, xr, xi, yr, yi, wr, wi, C32, S32, Fi, -1.0f);
  for (int j = lane; j < 1024; j += 32) ps[j] = yr[j] * yr[j] + yi[j] * yi[j];
  __syncthreads();

  // ---- low-frequency mirror replica ----
  const int   kk  = (int)floorf(f0 / DELTA + 1.0f - 1e-4f) + 1; // #bins with freq < f0+DELTA
  const float x0r = f0 - (float)(kk - 1) * DELTA;
  for (int j = lane; j < 1024; j += 32) {
    const float freq = (float)j * DELTA;
    const float xiv  = fminf(fmaxf(freq, x0r), f0);
    const float p    = (xiv - x0r) / DELTA;
    int b = min(max((int)floorf(p), 0), kk - 1);
    const float frac = p - (float)b;
    const float yb = ps[min(max(kk - 1 - b, 0), 1023)];
    const float dy = (b == kk - 1) ? 0.0f : ps[min(max(kk - 2 - b, 0), 1023)] - yb;
    wr[j] = (freq < f0) ? (yb + dy * frac) : 0.0f;   // staged adds (wr is free here)
  }
  __syncthreads();
  for (int j = lane; j < 1024; j += 32) ps[j] += wr[j];
  __syncthreads();
  // mirror upper half: ps[513..1023] = ps[1..511] reversed
  for (int j = lane; j < 1024; j += 32) wr[j] = ps[j];
  __syncthreads();
  for (int j = lane; j < 1024; j += 32)
    if (j > HALFN) ps[j] = wr[1024 - j];
  __syncthreads();

  // ---- cumulative integral of concat(ps, ps) * DELTA (wave-wide scan) ----
  {
    float carry = 0.0f;
    for (int c = 0; c < 64; ++c) {
      const int i = c * 32 + lane;
      float v = ps[i & 1023] * DELTA;
      for (int off = 1; off < 32; off <<= 1) {
        const float nv = __shfl_up(v, off, 32);
        if (lane >= off) v += nv;
      }
      dsb[i] = carry + v;
      carry += __shfl(v, 31, 32);
    }
  }
  __syncthreads();

  // ---- f0-adaptive rectangular smoothing ----
  const float x0d = -FSr + DELTA * 0.5f;
  for (int i = lane; i <= HALFN; i += 32) {
    const float cf = (float)i * DELTA;
    const float lo = interp_ds(dsb, x0d, cf - f0 / 3.0f);
    const float hi = interp_ds(dsb, x0d, cf + f0 / 3.0f);
    smb[i] = (hi - lo) * 1.5f / f0 + 1e-10f;
  }
  __syncthreads();
  for (int j = lane; j < 1024; j += 32) {
    const int idx = (j <= HALFN) ? j : (1024 - j);
    xr[j] = logf(smb[idx]);
    xi[j] = 0.0f;
  }
  __syncthreads();

  // ---- FFT #2: cepstrum + sinc lifter ----
  fft1024(lane, xr, xi, yr, yi, wr, wi, C32, S32, Fi, -1.0f);
  for (int j = lane; j < 1024; j += 32) {
    const float q = (float)j / FSr;
    const float a = PI_F * f0 * q;
    const float sl = (j == 0) ? 1.0f : (sinf(a) / a);
    const float cl = 1.0f - 2.0f * Q1C + 2.0f * Q1C * cosf(2.0f * a);
    const float g = sl * cl;
    yr[j] *= g; yi[j] *= g;
  }
  __syncthreads();

  // ---- FFT #3 (inverse direction) -> envelope ----
  fft1024(lane, yr, yi, xr, xi, wr, wi, C32, S32, Fi, +1.0f);
  for (int i = lane; i <= HALFN; i += 32)
    out[i * NFRAMES + frame] = expf(xr[i] * (1.0f / 1024.0f));
}

extern "C" void kernel_launch(void* const* d_in, const int* in_sizes, int n_in,
                              void* d_out, int out_size, void* d_ws, size_t ws_size,
                              hipStream_t stream) {
  const float* x   = (const float*)d_in[0];
  const float* f0  = (const float*)d_in[1];
  const float* tp  = (const float*)d_in[2];
  float* out = (float*)d_out;
  (void)in_sizes; (void)n_in; (void)out_size; (void)d_ws; (void)ws_size;
  cheaptrick_kernel<<<NFRAMES, 32, 0, stream>>>(x, f0, tp, out);
}